// NeuralDictionaryV8Double_38594576121952
// MI455X (gfx1250) — compile-verified
//
#include <hip/hip_runtime.h>
#include <math.h>

// ---------------------------------------------------------------------------
// Neural dictionary attention (N=200000, 128 features):
//   d[n]   = -sum_j |K[n,j] - q[j]|
//   w      = softmax(d)
//   out[j] = sum_n w[n] * V[n,j]
//
// HBM-bound: 205 MB @ 23.3 TB/s => ~8.8us floor. Single streaming pass with
// online (flash-style) softmax. Value accumulation on v_wmma_f32_16x16x4_f32
// (f32 WMMA => numerics identical to scalar FMA). This revision maximizes
// per-wave memory-level parallelism: all 16 key b128 loads + 64 value b32
// loads of a chunk are issued before any consumption, and the lane-reduction
// uses half-wave row pairs (40 ds ops/chunk instead of 80).
// ---------------------------------------------------------------------------

typedef float v2f __attribute__((ext_vector_type(2)));
typedef float v8f __attribute__((ext_vector_type(8)));

#define IN_F   128
#define OUT_F  128
#define NWG    512
#define WPW    8              // waves per workgroup
#define THREADS (WPW * 32)
#define CHUNK  16             // rows per wave-chunk (one WMMA batch)

__global__ __launch_bounds__(THREADS)
void nd_partial_kernel(const float* __restrict__ query,
                       const float* __restrict__ keys,
                       const float* __restrict__ values,
                       float* __restrict__ partials,   // [NWG][130] : m, s, acc[128]
                       int N)
{
    __shared__ float lds[WPW * 132];

    const int tid  = threadIdx.x;
    const int lane = tid & 31;
    const int wave = tid >> 5;
    const int half = lane >> 4;       // 0: rows 2p, 1: rows 2p+1
    const int hl   = lane & 15;       // lane within half-wave
    const bool lo  = (lane < 16);     // WMMA K-half selector
    const int col  = lane & 15;       // WMMA N column within a 16-wide tile

    // Query slice for the pair scheme: 8 contiguous features per lane.
    const float4 qa = *reinterpret_cast<const float4*>(query + 8 * hl);
    const float4 qb = *reinterpret_cast<const float4*>(query + 8 * hl + 4);

    const int nChunks    = (N + CHUNK - 1) / CHUNK;
    const int totalWaves = NWG * WPW;
    const int gw         = blockIdx.x * WPW + wave;

    float m = -INFINITY;   // running max logit
    float s = 0.f;         // running exp-sum
    v8f acc[8] = {};       // 8 j-tiles; all-equal-rows WMMA accumulators

    for (int c = gw; c < nChunks; c += totalWaves) {
        const int base = c * CHUNK;

        // ================= issue ALL loads up front =================
        // Keys: pair p -> half 0 reads row base+2p, half 1 reads base+2p+1;
        // each lane grabs 8 contiguous floats (512B contiguous per half-wave).
        float4 ka[8], kb[8];
#pragma unroll
        for (int p = 0; p < 8; ++p) {
            int row = base + 2 * p + half;
            int rc  = (row < N) ? row : (N - 1);
            const float* kr = keys + (size_t)rc * IN_F + 8 * hl;
            ka[p] = *reinterpret_cast<const float4*>(kr);
            kb[p] = *reinterpret_cast<const float4*>(kr + 4);
        }

        // Values (WMMA B operands): B layout => VGPR0 = K0(lo)/K2(hi) rows,
        // VGPR1 = K1/K3 rows, N = lane&15.
        float bx[32], by[32];
#pragma unroll
        for (int g = 0; g < 4; ++g) {
            const int r0  = base + 4 * g + (lo ? 0 : 2);
            const int r0c = (r0 < N) ? r0 : (N - 1);
            const int r1c = (r0 + 1 < N) ? (r0 + 1) : (N - 1);
            const float* v0 = values + (size_t)r0c * OUT_F;
            const float* v1 = values + (size_t)r1c * OUT_F;
#pragma unroll
            for (int t = 0; t < 8; ++t) {
                bx[g * 8 + t] = v0[t * 16 + col];
                by[g * 8 + t] = v1[t * 16 + col];
            }
        }

        // Prefetch the next chunk while this one is consumed.
        const int nc = c + totalWaves;
        if (nc < nChunks) {
            const char* kp = (const char*)(keys   + (size_t)nc * CHUNK * IN_F)  + lane * 256;
            const char* vp = (const char*)(values + (size_t)nc * CHUNK * OUT_F) + lane * 256;
            __builtin_prefetch(kp, 0, 0);
            __builtin_prefetch(vp, 0, 0);
        }

        // ================= distances (half-wave pair reduction) ==========
        float d[16];
#pragma unroll
        for (int p = 0; p < 8; ++p) {
            float r = fabsf(ka[p].x - qa.x) + fabsf(ka[p].y - qa.y) +
                      fabsf(ka[p].z - qa.z) + fabsf(ka[p].w - qa.w) +
                      fabsf(kb[p].x - qb.x) + fabsf(kb[p].y - qb.y) +
                      fabsf(kb[p].z - qb.z) + fabsf(kb[p].w - qb.w);
#pragma unroll
            for (int off = 8; off >= 1; off >>= 1)   // reduce within half-wave
                r += __shfl_xor(r, off, 32);
            const float o  = __shfl_xor(r, 16, 32);  // exchange across halves
            const float de = half ? o : r;           // row base+2p
            const float dd = half ? r : o;           // row base+2p+1
            d[2 * p + 0] = (base + 2 * p + 0 < N) ? -de : -INFINITY;
            d[2 * p + 1] = (base + 2 * p + 1 < N) ? -dd : -INFINITY;
        }

        // ================= online softmax update =================
        float cmax = d[0];
#pragma unroll
        for (int r = 1; r < 16; ++r) cmax = fmaxf(cmax, d[r]);
        const float newm  = fmaxf(m, cmax);
        const float scale = __expf(m - newm);        // m=-inf first time -> 0

        float w[16];
        float wsum = 0.f;
#pragma unroll
        for (int r = 0; r < 16; ++r) { w[r] = __expf(d[r] - newm); wsum += w[r]; }
        s = s * scale + wsum;
        m = newm;
#pragma unroll
        for (int t = 0; t < 8; ++t) acc[t] *= scale;

        // ================= 32 WMMAs, B already in registers ==============
        // A layout: lanes0-15 K={0,1}, lanes16-31 K={2,3}; weights replicated
        // across all 16 A rows so every D row holds the same accumulator.
#pragma unroll
        for (int g = 0; g < 4; ++g) {
            v2f a;
            a.x = lo ? w[4 * g + 0] : w[4 * g + 2];
            a.y = lo ? w[4 * g + 1] : w[4 * g + 3];
#pragma unroll
            for (int t = 0; t < 8; ++t) {
                v2f b;
                b.x = bx[g * 8 + t];
                b.y = by[g * 8 + t];
                acc[t] = __builtin_amdgcn_wmma_f32_16x16x4_f32(
                    false, a, false, b, (short)0, acc[t], false, false);
            }
        }
    }

    // ---- per-wave partial -> LDS ----
    float* myl = lds + wave * 132;
    if (lane == 0) { myl[0] = m; myl[1] = s; }
#pragma unroll
    for (int t = 0; t < 8; ++t)
        if (lo) myl[2 + t * 16 + lane] = acc[t][0];   // row 0 of D = the answer
    __syncthreads();

    // ---- combine 8 waves -> block partial ----
    float M = -INFINITY;
#pragma unroll
    for (int wv = 0; wv < WPW; ++wv) M = fmaxf(M, lds[wv * 132]);

    float* outp = partials + (size_t)blockIdx.x * 130;
    if (tid < 128) {
        float a = 0.f;
#pragma unroll
        for (int wv = 0; wv < WPW; ++wv) {
            const float mw = lds[wv * 132];
            const float e  = (mw == -INFINITY) ? 0.f : __expf(mw - M);
            a += lds[wv * 132 + 2 + tid] * e;
        }
        outp[2 + tid] = a;
    }
    if (tid == 0) {
        float stot = 0.f;
#pragma unroll
        for (int wv = 0; wv < WPW; ++wv) {
            const float mw = lds[wv * 132];
            const float e  = (mw == -INFINITY) ? 0.f : __expf(mw - M);
            stot += lds[wv * 132 + 1] * e;
        }
        outp[0] = M;
        outp[1] = stot;
    }
}

__global__ __launch_bounds__(128)
void nd_finalize_kernel(const float* __restrict__ partials,
                        float* __restrict__ out, int nParts)
{
    const int tid = threadIdx.x;   // 0..127 -> output feature
    float M = -INFINITY;
    for (int b = 0; b < nParts; ++b) M = fmaxf(M, partials[(size_t)b * 130]);
    float stot = 0.f;
    float a    = 0.f;
    for (int b = 0; b < nParts; ++b) {
        const float mb = partials[(size_t)b * 130];
        const float e  = (mb == -INFINITY) ? 0.f : __expf(mb - M);
        stot += partials[(size_t)b * 130 + 1] * e;
        a    += partials[(size_t)b * 130 + 2 + tid] * e;
    }
    out[tid] = a / stot;
}

extern "C" void kernel_launch(void* const* d_in, const int* in_sizes, int n_in,
                              void* d_out, int out_size, void* d_ws, size_t ws_size,
                              hipStream_t stream)
{
    const float* query  = (const float*)d_in[0];
    const float* keys   = (const float*)d_in[1];
    const float* values = (const float*)d_in[2];
    float* out          = (float*)d_out;
    float* partials     = (float*)d_ws;       // needs NWG*130*4 = ~266 KB

    const int N = in_sizes[1] / IN_F;

    nd_partial_kernel<<<NWG, THREADS, 0, stream>>>(query, keys, values, partials, N);
    nd_finalize_kernel<<<1, 128, 0, stream>>>(partials, out, NWG);
}